// Snail_69217692942422
// MI455X (gfx1250) — compile-verified
//
#include <hip/hip_runtime.h>
#include <hip/hip_bf16.h>
#include <math.h>

// ---------------------------------------------------------------------------
// Model dims
// ---------------------------------------------------------------------------
#define EPISODES 128
#define SEQ      26
#define RROWS    (EPISODES * SEQ)   // 3328
#define IN_CH    19
#define OUT_CH   64
#define HW0      22
#define TC_CH    128
#define CH0      69     // 64 + 5
#define CH1      101    // CH0 + 32
#define CH2      741    // CH1 + 5*128
#define CH3      805    // CH2 + 64
#define CH4      1445   // CH3 + 5*128
#define CH5      1573   // CH4 + 128

typedef float v2f __attribute__((ext_vector_type(2)));
typedef float v8f __attribute__((ext_vector_type(8)));

// fp32 WMMA: D(16x16) += A(16x4) * B(4x16), full wave32 op.
__device__ __forceinline__ v8f wmma_f32(v2f a, v2f b, v8f c) {
  return __builtin_amdgcn_wmma_f32_16x16x4_f32(false, a, false, b, (short)0, c,
                                               false, false);
}

#define V8F_ZERO {0.f, 0.f, 0.f, 0.f, 0.f, 0.f, 0.f, 0.f}

// ---------------------------------------------------------------------------
// GEMM: Out[m, ocol+n] = sum_k A[m,k]*Wt[n,k] + bias[n]
// Register-blocked: wave = 32x32 macro-tile (2 A frags x 2 B frags = 4 WMMA
// per K-step), block = 4 waves arranged 2x2 -> 64x64 per block.
// All loads unconditional from clamped row pointers; K tail peeled + masked.
// ---------------------------------------------------------------------------
__global__ void gemm_wmma(const float* __restrict__ A, int lda,
                          const float* __restrict__ Wt,
                          const float* __restrict__ bias,
                          float* __restrict__ Out, int ldo, int ocol,
                          int M, int N, int K) {
  const int lane = threadIdx.x & 31;
  const int wave = threadIdx.x >> 5;
  const int mw = wave & 1, nw = wave >> 1;
  const int lm = lane & 15;
  const int kh = (lane >> 4) << 1;  // 0 | 2
  const int m0 = blockIdx.x * 64 + mw * 32;
  const int n0 = blockIdx.y * 64 + nw * 32;

  auto rowA = [&](int m) {
    const int mm = (m < M) ? m : (M - 1);
    return A + (size_t)mm * lda + kh;
  };
  auto rowW = [&](int n) {
    const int nn = (n < N) ? n : (N - 1);
    return Wt + (size_t)nn * K + kh;
  };
  const float* __restrict__ pa0 = rowA(m0 + lm);
  const float* __restrict__ pa1 = rowA(m0 + 16 + lm);
  const float* __restrict__ pw0 = rowW(n0 + lm);
  const float* __restrict__ pw1 = rowW(n0 + 16 + lm);

  v8f c00 = V8F_ZERO, c01 = V8F_ZERO, c10 = V8F_ZERO, c11 = V8F_ZERO;
  const int kfull = K >> 2;
#pragma unroll 2
  for (int kk = 0; kk < kfull; ++kk) {
    v2f a0{pa0[0], pa0[1]}, a1{pa1[0], pa1[1]};
    v2f b0{pw0[0], pw0[1]}, b1{pw1[0], pw1[1]};
    pa0 += 4; pa1 += 4; pw0 += 4; pw1 += 4;
    c00 = wmma_f32(a0, b0, c00);
    c01 = wmma_f32(a0, b1, c01);
    c10 = wmma_f32(a1, b0, c10);
    c11 = wmma_f32(a1, b1, c11);
  }
  if (K & 3) {  // masked tail step (indices clamped, A masked -> product 0)
    const int k = (kfull << 2) + kh;
    const float q0 = (k < K) ? 1.f : 0.f;
    const float q1 = (k + 1 < K) ? 1.f : 0.f;
    const int d0 = (k < K) ? 0 : (k - (K - 1));
    const int d1 = (k + 1 < K) ? 1 : (K - 1 - k + 2 * (int)(k + 1 - K) * 0);
    // simpler: clamp absolute indices
    const int ks0 = (k < K) ? k : (K - 1);
    const int ks1 = (k + 1 < K) ? (k + 1) : (K - 1);
    const int o0 = ks0 - k, o1 = ks1 - k;
    (void)d0; (void)d1;
    v2f a0{q0 * pa0[o0], q1 * pa0[o1]};
    v2f a1{q0 * pa1[o0], q1 * pa1[o1]};
    v2f b0{pw0[o0], pw0[o1]};
    v2f b1{pw1[o0], pw1[o1]};
    c00 = wmma_f32(a0, b0, c00);
    c01 = wmma_f32(a0, b1, c01);
    c10 = wmma_f32(a1, b0, c10);
    c11 = wmma_f32(a1, b1, c11);
  }

  // D layout: VGPR r -> M = r (+8 for lanes 16-31); N = lane & 15
  const int rr8 = (lane >> 4) << 3;
  v8f accs[2][2] = {{c00, c01}, {c10, c11}};
#pragma unroll
  for (int ti = 0; ti < 2; ++ti) {
#pragma unroll
    for (int tj = 0; tj < 2; ++tj) {
      const int nc = n0 + tj * 16 + lm;
      if (nc < N) {
        const float bb = bias ? bias[nc] : 0.f;
#pragma unroll
        for (int r8 = 0; r8 < 8; ++r8) {
          const int mr = m0 + ti * 16 + rr8 + r8;
          if (mr < M) Out[(size_t)mr * ldo + ocol + nc] = accs[ti][tj][r8] + bb;
        }
      }
    }
  }
}

// ---------------------------------------------------------------------------
// conv3x3 same-pad, implicit GEMM. One wave owns ALL 64 output channels for a
// 16-position tile (one B gather feeds 4 WMMAs). 4 waves = 64 positions/block.
// K order is (off, ic): for each of the 9 taps validity is uniform, so inside
// the ic loop all loads are unconditional strided reads (mask folded as 0/1).
// ---------------------------------------------------------------------------
__global__ void conv3x3_wmma(const float* __restrict__ X,
                             const float* __restrict__ Wgt,  // [64][IC][3][3]
                             const float* __restrict__ bias, // [64]
                             float* __restrict__ Out,        // [R][64][hw][hw]
                             int R, int IC, int hw) {
  const int OC = 64;
  const int lane = threadIdx.x & 31;
  const int wave = threadIdx.x >> 5;
  const int lm = lane & 15;
  const int kh = (lane >> 4) << 1;
  const int P = hw * hw;
  const long Ntot = (long)R * P;
  const long n = (long)blockIdx.x * 64 + wave * 16 + lm;
  const bool nok = n < Ntot;
  const long ns = nok ? n : (Ntot - 1);
  const int r = (int)(ns / P);
  const int p = (int)(ns % P);
  const int oy = p / hw, ox = p % hw;
  const float* __restrict__ xr = X + (size_t)r * IC * P;
  const int wstride = IC * 9;

  v8f acc[4] = {V8F_ZERO, V8F_ZERO, V8F_ZERO, V8F_ZERO};
  const int icfull = IC >> 2;

  for (int off = 0; off < 9; ++off) {
    const int iy = oy + off / 3 - 1;
    const int ix = ox + off % 3 - 1;
    const bool valid = nok && iy >= 0 && iy < hw && ix >= 0 && ix < hw;
    const float vm = valid ? 1.f : 0.f;
    const int iyc = iy < 0 ? 0 : (iy >= hw ? hw - 1 : iy);
    const int ixc = ix < 0 ? 0 : (ix >= hw ? hw - 1 : ix);
    const float* __restrict__ bp = xr + iyc * hw + ixc + (size_t)kh * P;
    const float* __restrict__ ap = Wgt + (size_t)lm * wstride + off + kh * 9;
#pragma unroll 2
    for (int ii = 0; ii < icfull; ++ii) {
      v2f b{vm * bp[0], vm * bp[(size_t)P]};
#pragma unroll
      for (int t = 0; t < 4; ++t) {
        const float* at = ap + (size_t)t * 16 * wstride;
        v2f a{at[0], at[9]};
        acc[t] = wmma_f32(a, b, acc[t]);
      }
      bp += (size_t)4 * P;
      ap += 36;
    }
    if (IC & 3) {  // masked ic tail
      const int ic = (icfull << 2) + kh;
      const float q0 = (ic < IC) ? vm : 0.f;
      const float q1 = (ic + 1 < IC) ? vm : 0.f;
      const long s0 = (ic < IC) ? 0 : -(long)P;      // clamp back into range
      const long s1 = (ic + 1 < IC) ? (long)P : 0;
      const long w0 = (ic < IC) ? 0 : -9;
      const long w1 = (ic + 1 < IC) ? 9 : 0;
      v2f b{q0 * bp[s0], q1 * bp[s1]};
#pragma unroll
      for (int t = 0; t < 4; ++t) {
        const float* at = ap + (size_t)t * 16 * wstride;
        v2f a{at[w0], at[w1]};
        acc[t] = wmma_f32(a, b, acc[t]);
      }
    }
  }

  const int rr8 = (lane >> 4) << 3;
  if (nok) {
#pragma unroll
    for (int t = 0; t < 4; ++t) {
#pragma unroll
      for (int r8 = 0; r8 < 8; ++r8) {
        const int oc = t * 16 + rr8 + r8;
        Out[((size_t)r * OC + oc) * P + p] = acc[t][r8] + bias[oc];
      }
    }
  }
}

// ---------------------------------------------------------------------------
// Per-channel batch stats over (N,H,W); biased variance (training-mode BN).
// ---------------------------------------------------------------------------
__global__ void bn_stats(const float* __restrict__ X, int R, int P,
                         float* __restrict__ mean, float* __restrict__ invstd) {
  const int C = 64;
  const int ch = blockIdx.x;
  const long total = (long)R * P;
  float s = 0.f, sq = 0.f;
  for (long i = threadIdx.x; i < total; i += blockDim.x) {
    const long rr = i / P;
    const int pp = (int)(i % P);
    const float v = X[((size_t)rr * C + ch) * P + pp];
    s += v;
    sq += v * v;
  }
  __shared__ float sh0[256], sh1[256];
  sh0[threadIdx.x] = s;
  sh1[threadIdx.x] = sq;
  __syncthreads();
  for (int off = 128; off > 0; off >>= 1) {
    if ((int)threadIdx.x < off) {
      sh0[threadIdx.x] += sh0[threadIdx.x + off];
      sh1[threadIdx.x] += sh1[threadIdx.x + off];
    }
    __syncthreads();
  }
  if (threadIdx.x == 0) {
    const float m = sh0[0] / (float)total;
    const float v = sh1[0] / (float)total - m * m;
    mean[ch] = m;
    invstd[ch] = rsqrtf(v + 1e-5f);
  }
}

// ---------------------------------------------------------------------------
// Fused BN (affine) + ReLU + 2x2 maxpool (VALID).
// ---------------------------------------------------------------------------
__global__ void bn_relu_pool(const float* __restrict__ X, float* __restrict__ Out,
                             const float* __restrict__ mean,
                             const float* __restrict__ invstd,
                             const float* __restrict__ gamma,
                             const float* __restrict__ beta, int R, int hw) {
  const int C = 64;
  const int ho = hw >> 1;
  const long tot = (long)R * C * ho * ho;
  const long i = (long)blockIdx.x * blockDim.x + threadIdx.x;
  if (i >= tot) return;
  const int px = (int)(i % ho);
  long t = i / ho;
  const int py = (int)(t % ho);
  t /= ho;
  const int c = (int)(t % C);
  const int rr = (int)(t / C);
  const float g = gamma[c] * invstd[c];
  const float bb = beta[c] - mean[c] * g;
  const float* xp = X + ((size_t)rr * C + c) * hw * hw;
  float mx = -3.4e38f;
#pragma unroll
  for (int dy = 0; dy < 2; ++dy)
#pragma unroll
    for (int dx = 0; dx < 2; ++dx) {
      const float v = fmaxf(g * xp[(2 * py + dy) * hw + (2 * px + dx)] + bb, 0.f);
      mx = fmaxf(mx, v);
    }
  Out[((size_t)rr * C + c) * ho * ho + py * ho + px] = mx;
}

__global__ void concat_h0(const float* __restrict__ emb, const float* __restrict__ y,
                          float* __restrict__ dst, int R, int ld) {
  const long tot = (long)R * CH0;
  const long i = (long)blockIdx.x * blockDim.x + threadIdx.x;
  if (i >= tot) return;
  const int ch = (int)(i % CH0);
  const int rr = (int)(i / CH0);
  dst[(size_t)rr * ld + ch] =
      (ch < OUT_CH) ? emb[(size_t)rr * OUT_CH + ch] : y[(size_t)rr * 5 + (ch - OUT_CH)];
}

__global__ void copy_ch(const float* __restrict__ src, int lds,
                        float* __restrict__ dst, int ldd, int R, int C) {
  const long tot = (long)R * C;
  const long i = (long)blockIdx.x * blockDim.x + threadIdx.x;
  if (i >= tot) return;
  const int ch = (int)(i % C);
  const int rr = (int)(i / C);
  dst[(size_t)rr * ldd + ch] = src[(size_t)rr * lds + ch];
}

// ---------------------------------------------------------------------------
// Attention mixing: scores = QK^T, causal mask (k>q -> -inf), softmax over the
// QUERY axis (dim=1, as the reference does), out = attn @ V at column `ocol`.
// ---------------------------------------------------------------------------
__global__ void attn_mix(const float* __restrict__ Q, const float* __restrict__ Kt,
                         const float* __restrict__ V, float* __restrict__ Dst,
                         int ldd, int ocol, int ks, int vs) {
  const int S = SEQ;
  const int ep = blockIdx.x;
  __shared__ float sc[SEQ][SEQ];
  const float scale = rsqrtf((float)ks);
  for (int pidx = threadIdx.x; pidx < S * S; pidx += blockDim.x) {
    const int q = pidx / S, k = pidx % S;
    float d;
    if (k <= q) {
      const float* qp = Q + ((size_t)ep * S + q) * ks;
      const float* kp = Kt + ((size_t)ep * S + k) * ks;
      float acc = 0.f;
      for (int j = 0; j < ks; ++j) acc += qp[j] * kp[j];
      d = acc * scale;
    } else {
      d = -3.0e38f;
    }
    sc[q][k] = d;
  }
  __syncthreads();
  if ((int)threadIdx.x < S) {
    const int k = threadIdx.x;  // softmax over q for this column
    float mx = -3.4e38f;
    for (int q = k; q < S; ++q) mx = fmaxf(mx, sc[q][k]);
    float sum = 0.f;
    for (int q = k; q < S; ++q) {
      const float e = __expf(sc[q][k] - mx);
      sc[q][k] = e;
      sum += e;
    }
    const float inv = 1.f / sum;
    for (int q = 0; q < S; ++q) sc[q][k] = (q >= k) ? sc[q][k] * inv : 0.f;
  }
  __syncthreads();
  for (int pidx = threadIdx.x; pidx < S * vs; pidx += blockDim.x) {
    const int q = pidx / vs, d = pidx % vs;
    float acc = 0.f;
    for (int k = 0; k < S; ++k) acc += sc[q][k] * V[((size_t)ep * S + k) * vs + d];
    Dst[(size_t)(ep * S + q) * ldd + ocol + d] = acc;
  }
}

// ---------------------------------------------------------------------------
// One TC layer: dual implicit GEMM (gate & feature) x 2 N-tiles share one A
// fragment (4 WMMA / K-step). K split into past-tap half (mask = t>=dil as
// 0/1 float) and current-tap half; both branch-free with peeled masked tails.
// Wave = 16M x 32N; block 2x2 waves = 32M x 64N; grid.y=2 covers 128 ch.
// ---------------------------------------------------------------------------
__global__ void tc_layer_wmma(const float* __restrict__ H, int ldh,
                              const float* __restrict__ W1, const float* __restrict__ B1,
                              const float* __restrict__ W2, const float* __restrict__ B2,
                              float* __restrict__ Dst, int ldd, int ocol,
                              int M, int cin, int dil) {
  const int lane = threadIdx.x & 31;
  const int wave = threadIdx.x >> 5;
  const int mw = wave & 1, nw = wave >> 1;
  const int lm = lane & 15;
  const int kh = (lane >> 4) << 1;
  const int m0 = blockIdx.x * 32 + mw * 16;
  const int n0 = blockIdx.y * 64 + nw * 32;

  const int m = m0 + lm;  // M == RROWS divides 32, always in range
  const int t = m % SEQ;
  const bool pastok = t >= dil;
  const float pm = pastok ? 1.f : 0.f;
  const float* __restrict__ hp = H + (size_t)(pastok ? (m - dil) : m) * ldh + kh;
  const float* __restrict__ hc = H + (size_t)m * ldh + kh;
  const size_t wst = (size_t)cin * 2;
  const float* __restrict__ w1r0 = W1 + (size_t)(n0 + lm) * wst + 2 * kh;
  const float* __restrict__ w1r1 = W1 + (size_t)(n0 + 16 + lm) * wst + 2 * kh;
  const float* __restrict__ w2r0 = W2 + (size_t)(n0 + lm) * wst + 2 * kh;
  const float* __restrict__ w2r1 = W2 + (size_t)(n0 + 16 + lm) * wst + 2 * kh;

  v8f cg0 = V8F_ZERO, cg1 = V8F_ZERO, cf0 = V8F_ZERO, cf1 = V8F_ZERO;
  const int cfull = cin >> 2;
  const int ctail = cin & 3;
  const int ictail = (cfull << 2) + kh;
  const long to0 = (ictail < cin) ? 0 : -1;         // clamped tail offsets
  const long to1 = (ictail + 1 < cin) ? 1 : to0;
  const float tq0 = (ictail < cin) ? 1.f : 0.f;
  const float tq1 = (ictail + 1 < cin) ? 1.f : 0.f;

  // ---- half 0: past tap (tap index 0), A masked by pm ----
  {
    const float* ha = hp;
    const float* p10 = w1r0, * p11 = w1r1, * p20 = w2r0, * p21 = w2r1;
#pragma unroll 2
    for (int ii = 0; ii < cfull; ++ii) {
      v2f a{pm * ha[0], pm * ha[1]};
      v2f b10{p10[0], p10[2]}, b11{p11[0], p11[2]};
      v2f b20{p20[0], p20[2]}, b21{p21[0], p21[2]};
      ha += 4; p10 += 8; p11 += 8; p20 += 8; p21 += 8;
      cg0 = wmma_f32(a, b10, cg0);
      cg1 = wmma_f32(a, b11, cg1);
      cf0 = wmma_f32(a, b20, cf0);
      cf1 = wmma_f32(a, b21, cf1);
    }
    if (ctail) {
      v2f a{tq0 * pm * ha[to0], tq1 * pm * ha[to1]};
      v2f b10{p10[2 * to0], p10[2 * to1]}, b11{p11[2 * to0], p11[2 * to1]};
      v2f b20{p20[2 * to0], p20[2 * to1]}, b21{p21[2 * to0], p21[2 * to1]};
      cg0 = wmma_f32(a, b10, cg0);
      cg1 = wmma_f32(a, b11, cg1);
      cf0 = wmma_f32(a, b20, cf0);
      cf1 = wmma_f32(a, b21, cf1);
    }
  }
  // ---- half 1: current tap (tap index 1) ----
  {
    const float* ha = hc;
    const float* p10 = w1r0 + 1, * p11 = w1r1 + 1, * p20 = w2r0 + 1, * p21 = w2r1 + 1;
#pragma unroll 2
    for (int ii = 0; ii < cfull; ++ii) {
      v2f a{ha[0], ha[1]};
      v2f b10{p10[0], p10[2]}, b11{p11[0], p11[2]};
      v2f b20{p20[0], p20[2]}, b21{p21[0], p21[2]};
      ha += 4; p10 += 8; p11 += 8; p20 += 8; p21 += 8;
      cg0 = wmma_f32(a, b10, cg0);
      cg1 = wmma_f32(a, b11, cg1);
      cf0 = wmma_f32(a, b20, cf0);
      cf1 = wmma_f32(a, b21, cf1);
    }
    if (ctail) {
      v2f a{tq0 * ha[to0], tq1 * ha[to1]};
      v2f b10{p10[2 * to0], p10[2 * to1]}, b11{p11[2 * to0], p11[2 * to1]};
      v2f b20{p20[2 * to0], p20[2 * to1]}, b21{p21[2 * to0], p21[2 * to1]};
      cg0 = wmma_f32(a, b10, cg0);
      cg1 = wmma_f32(a, b11, cg1);
      cf0 = wmma_f32(a, b20, cf0);
      cf1 = wmma_f32(a, b21, cf1);
    }
  }

  const int rr8 = (lane >> 4) << 3;
  v8f g2[2] = {cg0, cg1}, f2[2] = {cf0, cf1};
#pragma unroll
  for (int tj = 0; tj < 2; ++tj) {
    const int nc = n0 + tj * 16 + lm;
    const float bb1 = B1[nc], bb2 = B2[nc];
#pragma unroll
    for (int r8 = 0; r8 < 8; ++r8) {
      const int mr = m0 + rr8 + r8;
      const float g = g2[tj][r8] + bb1;
      const float f = f2[tj][r8] + bb2;
      const float gate = 1.f / (1.f + __expf(-g));
      Dst[(size_t)mr * ldd + ocol + nc] = gate * tanhf(f);
    }
  }
}

// ---------------------------------------------------------------------------
// Final: softmax over logits of the last sequence element of each episode.
// ---------------------------------------------------------------------------
__global__ void last_softmax(const float* __restrict__ logits, float* __restrict__ out) {
  const int b = blockIdx.x * blockDim.x + threadIdx.x;
  if (b >= EPISODES) return;
  const float* l = logits + ((size_t)b * SEQ + (SEQ - 1)) * 5;
  float mx = l[0];
#pragma unroll
  for (int i = 1; i < 5; ++i) mx = fmaxf(mx, l[i]);
  float e[5], s = 0.f;
#pragma unroll
  for (int i = 0; i < 5; ++i) {
    e[i] = __expf(l[i] - mx);
    s += e[i];
  }
  const float inv = 1.f / s;
#pragma unroll
  for (int i = 0; i < 5; ++i) out[(size_t)b * 5 + i] = e[i] * inv;
}

// ---------------------------------------------------------------------------
// Host orchestration
// ---------------------------------------------------------------------------
extern "C" void kernel_launch(void* const* d_in, const int* in_sizes, int n_in,
                              void* d_out, int out_size, void* d_ws, size_t ws_size,
                              hipStream_t stream) {
  (void)in_sizes; (void)n_in; (void)out_size; (void)ws_size;
  const int R = RROWS;
  auto F = [&](int i) -> const float* { return (const float*)d_in[i]; };

  // Leaf order: top-level (x, y, params); params flattened with sorted keys:
  //   attn1{bk,bq,bv,wk,wq,wv}=2..7  attn2=8..13  attn3=14..19
  //   embed[i]{b,beta,gamma,w}=20+4i  fc{b,w}=36,37
  //   tc1[i]{b1,b2,w1,w2}=38+4i       tc2[i]=58+4i
  const float* x = F(0);
  const float* y = F(1);

  char* wptr = (char*)d_ws;
  auto alloc = [&](size_t nfl) -> float* {
    float* p = (float*)wptr;
    wptr += ((nfl * sizeof(float) + 255) / 256) * 256;
    return p;
  };
  float* convA = alloc((size_t)R * 64 * HW0 * HW0);
  float* poolA = alloc((size_t)R * 64 * 11 * 11);
  float* meanb = alloc(64);
  float* istdb = alloc(64);
  float* SB1 = alloc((size_t)R * CH2);   // stride 741
  float* SB2 = alloc((size_t)R * CH4);   // stride 1445
  float* SB3 = alloc((size_t)R * CH5);   // stride 1573
  float* qb = alloc((size_t)R * 256);
  float* kb = alloc((size_t)R * 256);
  float* vb = alloc((size_t)R * 128);
  float* logitsb = alloc((size_t)R * 5);

  const dim3 blk128(128);

  auto conv_stage = [&](const float* in, int IC, int hw, int ei) {
    const float* cb  = F(20 + 4 * ei + 0);
    const float* bet = F(20 + 4 * ei + 1);
    const float* gam = F(20 + 4 * ei + 2);
    const float* wgt = F(20 + 4 * ei + 3);
    const long Ntot = (long)R * hw * hw;
    conv3x3_wmma<<<dim3((unsigned)((Ntot + 63) / 64)), blk128, 0, stream>>>(
        in, wgt, cb, convA, R, IC, hw);
    bn_stats<<<64, 256, 0, stream>>>(convA, R, hw * hw, meanb, istdb);
    const int ho = hw >> 1;
    const long tot = (long)R * 64 * ho * ho;
    bn_relu_pool<<<dim3((unsigned)((tot + 255) / 256)), 256, 0, stream>>>(
        convA, poolA, meanb, istdb, gam, bet, R, hw);
  };
  conv_stage(x, IN_CH, 22, 0);
  conv_stage(poolA, 64, 11, 1);
  conv_stage(poolA, 64, 5, 2);
  conv_stage(poolA, 64, 2, 3);

  {
    const long tot = (long)R * CH0;
    concat_h0<<<dim3((unsigned)((tot + 255) / 256)), 256, 0, stream>>>(poolA, y, SB1, R, CH2);
  }

  auto attention = [&](float* SB, int ld, int C, int ks, int vs, int abase, int ocol) {
    const float* bk = F(abase + 0);
    const float* bq = F(abase + 1);
    const float* bv = F(abase + 2);
    const float* wk = F(abase + 3);
    const float* wq = F(abase + 4);
    const float* wv = F(abase + 5);
    dim3 gq((R + 63) / 64, (ks + 63) / 64);
    gemm_wmma<<<gq, blk128, 0, stream>>>(SB, ld, wq, bq, qb, ks, 0, R, ks, C);
    gemm_wmma<<<gq, blk128, 0, stream>>>(SB, ld, wk, bk, kb, ks, 0, R, ks, C);
    dim3 gv((R + 63) / 64, (vs + 63) / 64);
    gemm_wmma<<<gv, blk128, 0, stream>>>(SB, ld, wv, bv, vb, vs, 0, R, vs, C);
    attn_mix<<<EPISODES, 256, 0, stream>>>(qb, kb, vb, SB, ld, ocol, ks, vs);
  };

  auto tcstack = [&](float* SB, int ld, int c0, int tbase) {
    for (int i = 0; i < 5; ++i) {
      const int cin = c0 + TC_CH * i;
      dim3 g(R / 32, 2);
      tc_layer_wmma<<<g, blk128, 0, stream>>>(
          SB, ld, F(tbase + 4 * i + 2), F(tbase + 4 * i + 0),
          F(tbase + 4 * i + 3), F(tbase + 4 * i + 1), SB, ld, cin, R, cin, 1 << i);
    }
  };

  attention(SB1, CH2, CH0, 64, 32, 2, CH0);    // attn1 -> SB1[:, 69:101]
  tcstack(SB1, CH2, CH1, 38);                  // tc1   -> SB1[:, 101:741]

  {
    const long tot = (long)R * CH2;
    copy_ch<<<dim3((unsigned)((tot + 255) / 256)), 256, 0, stream>>>(SB1, CH2, SB2, CH4, R, CH2);
  }
  attention(SB2, CH4, CH2, 128, 64, 8, CH2);   // attn2 -> SB2[:, 741:805]
  tcstack(SB2, CH4, CH3, 58);                  // tc2   -> SB2[:, 805:1445]

  {
    const long tot = (long)R * CH4;
    copy_ch<<<dim3((unsigned)((tot + 255) / 256)), 256, 0, stream>>>(SB2, CH4, SB3, CH5, R, CH4);
  }
  attention(SB3, CH5, CH4, 256, 128, 14, CH4); // attn3 -> SB3[:, 1445:1573]

  {
    dim3 gf((R + 63) / 64, 1);
    gemm_wmma<<<gf, blk128, 0, stream>>>(SB3, CH5, F(37), F(36), logitsb, 5, 0, R, 5, CH5);
    last_softmax<<<1, 128, 0, stream>>>(logitsb, (float*)d_out);
  }
}